// PhysicsForwardModel_59579786330213
// MI455X (gfx1250) — compile-verified
//
#include <hip/hip_runtime.h>
#include <math.h>

// ---------------------------------------------------------------------------
// PhysicsForwardModel on MI455X (gfx1250, wave32).
//   Lx = Ly = 512, Nx = Nz = 128, B = 2.
// Working set (~7MB) is L2-resident (192MB L2); runtime is dominated by the
// cos-scan stage (v_cos_f32). GEMMs use V_WMMA_F32_16X16X4_F32, one wave per
// 16x16 output tile, fragments loaded as b64 where K-contiguous.
// ---------------------------------------------------------------------------

typedef __attribute__((ext_vector_type(2))) float v2f;
typedef __attribute__((ext_vector_type(8))) float v8f;

#define LL 512      // Lx == Ly
#define NXX 128     // Nx == Nz
#define PI_F 3.14159265358979323846f

// --- Build orthonormal DCT-II matrix C[k][n], a[k] = C[k][0], and the
//     t-invariant frequency magnitude F[k][j] = sqrt(ky^2 + kx^2). ----------
__global__ void k_build_dct(float* __restrict__ C, float* __restrict__ avec,
                            float* __restrict__ F) {
  int idx = blockIdx.x * blockDim.x + threadIdx.x;   // 512*512 threads
  int k = idx >> 9;
  int n = idx & (LL - 1);
  float s = (k == 0) ? rsqrtf(4.0f * LL) : rsqrtf(2.0f * LL);
  float v = 2.0f * s * cosf(PI_F * (2.0f * (float)n + 1.0f) * (float)k / (2.0f * LL));
  C[idx] = v;
  if (n == 0) avec[k] = v;
  const float w = PI_F / (float)LL;
  float ky = (float)k * w;
  float kx = (float)n * w;
  F[idx] = sqrtf(ky * ky + kx * kx);
}

// --- Strided f32 WMMA GEMM: D[M x N] = alpha * rowScale[m] * (A @ B) --------
// A[m,k] at A[m*sa0 + k]          (K-stride 1 in all call sites -> b64 loads)
// B[k,n] at B[k*sb0 + n*sb1]; BCONTIG => sb0 == 1 (b64 loads for B too).
// One wave per 16x16 tile of D; K-loop in steps of 4 (V_WMMA_F32_16X16X4_F32).
template <bool BCONTIG>
__global__ void k_wmma_gemm(const float* __restrict__ A, int sa0,
                            const float* __restrict__ B, int sb0, int sb1,
                            float* __restrict__ D, int ldd,
                            int M, int N, int K,
                            float alpha, const float* __restrict__ rowScale) {
  int wave   = (blockIdx.x * blockDim.x + threadIdx.x) >> 5;
  int tilesN = N >> 4;
  int tilesM = M >> 4;
  if (wave >= tilesM * tilesN) return;           // uniform per wave: EXEC stays all-1s
  int tM = wave / tilesN;
  int tN = wave - tM * tilesN;
  int lane = threadIdx.x & 31;
  int half = lane >> 4;                          // 0: lanes 0-15, 1: lanes 16-31
  int l15  = lane & 15;
  int mBase = tM << 4, nBase = tN << 4;

  // ISA 32-bit A 16x4 layout: lanes0-15 -> K=kk,kk+1 ; lanes16-31 -> K=kk+2,kk+3
  const float* Arow = A + (size_t)(mBase + l15) * (size_t)sa0 + (size_t)(half << 1);
  const float* Bcol = B + (size_t)(nBase + l15) * (size_t)sb1 +
                      (size_t)(half << 1) * (size_t)(BCONTIG ? 1 : sb0);

  v8f acc = {};
  for (int kk = 0; kk < K; kk += 4) {
    v2f af = *(const v2f*)Arow;                  // one global_load_b64
    v2f bf;
    if (BCONTIG) {
      bf = *(const v2f*)Bcol;                    // one global_load_b64
      Bcol += 4;
    } else {
      bf[0] = Bcol[0];
      bf[1] = Bcol[(size_t)sb0];
      Bcol += (size_t)sb0 * 4;
    }
    Arow += 4;
    // (neg_a, A, neg_b, B, c_mod, C, reuse_a, reuse_b)
    acc = __builtin_amdgcn_wmma_f32_16x16x4_f32(
        false, af, false, bf, (short)0, acc, false, false);
  }

  // 32-bit C/D 16x16 layout: VGPR i -> row mBase + i + 8*half, col nBase + l15
#pragma unroll
  for (int i = 0; i < 8; ++i) {
    int row = mBase + i + (half << 3);
    float v = acc[i] * alpha;
    if (rowScale) v *= rowScale[row];
    D[(size_t)row * (size_t)ldd + (size_t)(nBase + l15)] = v;
  }
}

// --- Scan stage: V[t][b][j] = sum_k cos(F[k,j]*t) * M[b][k][j] --------------
// Dominant cost: 2*512^3 cos+fma on the TRANS/VALU pipes. F precomputed
// (t-invariant) so the loop is load,load,mul,cos,fma. Both streams are
// coalesced in j and L2-resident across the 512 t-values that reuse them.
__global__ void k_scan(const float* __restrict__ Mmat, const float* __restrict__ F,
                       float* __restrict__ V) {
  int idx = blockIdx.x * blockDim.x + threadIdx.x;   // 2*512*512
  int j = idx & (LL - 1);
  int t = (idx >> 9) & (LL - 1);
  int b = idx >> 18;
  float tf = (float)t;
  const float* Mb = Mmat + (size_t)b * LL * LL + j;
  const float* Fc = F + j;
  float acc = 0.0f;
#pragma unroll 8
  for (int k = 0; k < LL; ++k) {
    float ph = __cosf(Fc[(size_t)k * LL] * tf);
    acc = fmaf(ph, Mb[(size_t)k * LL], acc);
  }
  V[((size_t)t * 2 + (size_t)b) * LL + j] = acc;     // layout [t][b][j]
}

// --- Per-batch sum / sum-of-squares reduction (for std) ---------------------
__global__ void k_reduce(const float* __restrict__ P, float* __restrict__ acc4) {
  int b = blockIdx.y;
  const float* Pb = P + (size_t)b * LL * LL;
  float s = 0.0f, s2 = 0.0f;
  for (int i = blockIdx.x * blockDim.x + threadIdx.x; i < LL * LL;
       i += gridDim.x * blockDim.x) {
    float v = Pb[i];
    s += v;
    s2 += v * v;
  }
  __shared__ float sh[256], sh2[256];
  sh[threadIdx.x] = s;
  sh2[threadIdx.x] = s2;
  __syncthreads();
  for (int o = 128; o > 0; o >>= 1) {
    if (threadIdx.x < o) {
      sh[threadIdx.x]  += sh[threadIdx.x + o];
      sh2[threadIdx.x] += sh2[threadIdx.x + o];
    }
    __syncthreads();
  }
  if (threadIdx.x == 0) {
    atomicAdd(&acc4[2 * b + 0], sh[0]);
    atomicAdd(&acc4[2 * b + 1], sh2[0]);
  }
}

// --- Deterministic Gaussian noise (PCG hash + Box-Muller), sigma=1 ----------
__device__ __forceinline__ unsigned pcg_hash(unsigned v) {
  v = v * 747796405u + 2891336453u;
  unsigned w = ((v >> ((v >> 28) + 4u)) ^ v) * 277803737u;
  return (w >> 22) ^ w;
}

__global__ void k_noise(float* __restrict__ P, const float* __restrict__ acc4) {
  int idx = blockIdx.x * blockDim.x + threadIdx.x;   // 2*512*512
  int b = idx >> 18;
  const float inv_n = 1.0f / (float)(LL * LL);
  float mean = acc4[2 * b + 0] * inv_n;
  float var  = fmaxf(acc4[2 * b + 1] * inv_n - mean * mean, 0.0f);
  float stdv = sqrtf(var);
  unsigned h1 = pcg_hash(0x9E3779B9u ^ (unsigned)idx);
  unsigned h2 = pcg_hash(h1 + 0x85EBCA6Bu);
  float u1 = ((float)h1 + 1.0f) * 2.3283064365386963e-10f;  // (0,1]
  float u2 = (float)h2 * 2.3283064365386963e-10f;
  float g  = sqrtf(-2.0f * __logf(u1)) * __cosf(6.283185307179586f * u2);
  P[idx] += g * stdv;
}

// ---------------------------------------------------------------------------
extern "C" void kernel_launch(void* const* d_in, const int* in_sizes, int n_in,
                              void* d_out, int out_size, void* d_ws, size_t ws_size,
                              hipStream_t stream) {
  const float* x = (const float*)d_in[0];   // (2,1,128,128) f32; tau/chi unused
  float* out = (float*)d_out;               // (2,1,512,512) f32 = 524288

  // Workspace layout (floats). Total ~1,704,452 floats (~6.9 MB).
  float* ws   = (float*)d_ws;
  float* C    = ws;                                // 512*512
  float* avec = C + LL * LL;                       // 512
  float* F    = avec + LL;                         // 512*512
  float* U    = F + LL * LL;                       // 2 * 128*512
  float* Mm   = U + 2 * NXX * LL;                  // 2 * 512*512
  float* V    = Mm + 2 * LL * LL;                  // 512*2*512
  float* acc4 = V + 2 * LL * LL;                   // 4

  hipMemsetAsync(acc4, 0, 4 * sizeof(float), stream);

  // DCT matrix, a[k] = C[k][0], and frequency magnitudes F[k][j]
  k_build_dct<<<(LL * LL) / 256, 256, 0, stream>>>(C, avec, F);

  for (int b = 0; b < 2; ++b) {
    // U[b](128x512) = x[b](128x128) @ Bx,  Bx[j,l] = C[l][128+j]  (K=128)
    // B K-stride == 1 -> contiguous variant.
    int tiles = (NXX / 16) * (LL / 16);            // 256 tiles, 8 waves/block
    k_wmma_gemm<true><<<(tiles + 7) / 8, 256, 0, stream>>>(
        x + (size_t)b * NXX * NXX, NXX,
        C + NXX, 1, LL,
        U + (size_t)b * NXX * LL, LL,
        NXX, LL, NXX, 1.0f, nullptr);

    // Mm[b](512x512) = a[k] * ( C[:,0:128] @ U[b] )   (K=128)
    tiles = (LL / 16) * (LL / 16);                 // 1024 tiles
    k_wmma_gemm<false><<<(tiles + 7) / 8, 256, 0, stream>>>(
        C, LL,
        U + (size_t)b * NXX * LL, LL, 1,
        Mm + (size_t)b * LL * LL, LL,
        LL, LL, NXX, 1.0f, avec);
  }

  // V[t][b][j] = sum_k cos(F[k,j]*t) * Mm[b][k][j]
  k_scan<<<(2 * LL * LL) / 256, 256, 0, stream>>>(Mm, F, V);

  for (int b = 0; b < 2; ++b) {
    // Pdet[b](512x512) = (V[:,b,:] @ C) / 3   (K=512), straight into d_out
    int tiles = (LL / 16) * (LL / 16);
    k_wmma_gemm<false><<<(tiles + 7) / 8, 256, 0, stream>>>(
        V + (size_t)b * LL, 2 * LL,
        C, LL, 1,
        out + (size_t)b * LL * LL, LL,
        LL, LL, LL, (1.0f / 3.0f), nullptr);
  }

  // Per-batch std, then additive Gaussian noise (sigma = 1)
  dim3 rgrid(64, 2);
  k_reduce<<<rgrid, 256, 0, stream>>>(out, acc4);
  k_noise<<<(2 * LL * LL) / 256, 256, 0, stream>>>(out, acc4);
}